// VQQuantizer_84258668413389
// MI455X (gfx1250) — compile-verified
//
#include <hip/hip_runtime.h>

typedef __attribute__((ext_vector_type(2))) float v2f;
typedef __attribute__((ext_vector_type(8))) float v8f;

#define B_TOTAL 262144
#define D_DIM   64
#define K_CODES 1024
#define BETA_W  0.25f
#define DECAY_W 0.99f
#define EPS_W   1e-5f

#define LDS_STRIDE 68      // dwords; float4-aligned (68*4=272=16*17), conflict-free b64/b128 access
#define EROWS      64      // embedding rows per staged chunk
#define NSTAGES    (K_CODES / EROWS)   // 16
#define WAVES      8       // 256 threads / wave32

// ---------------- ws zero init (cluster_size + embed_sum + loss) ----------------
__global__ void vq_zero(float* ws, int count) {
    int i = blockIdx.x * blockDim.x + threadIdx.x;
    if (i < count) ws[i] = 0.0f;
}

// ---------------- per-code squared norms ----------------
__global__ void vq_enorm(const float* __restrict__ E, float* __restrict__ enorm) {
    int k = blockIdx.x * blockDim.x + threadIdx.x;
    if (k >= K_CODES) return;
    const float4* row = (const float4*)&E[(size_t)k * D_DIM];
    float s = 0.0f;
#pragma unroll
    for (int i = 0; i < D_DIM / 4; ++i) {
        float4 v = row[i];
        s += v.x * v.x + v.y * v.y + v.z * v.z + v.w * v.w;
    }
    enorm[k] = s;
}

// ---------------- main fused kernel: async-staged GEMM + argmin + gather + scatter ----------------
__global__ __launch_bounds__(256) void vq_main(
        const float* __restrict__ x, const float* __restrict__ E,
        const float* __restrict__ enorm,
        float* __restrict__ out_codes, float* __restrict__ out_q,
        float* __restrict__ out_res,
        float* ws_cluster, float* ws_esum, float* ws_loss)
{
    __shared__ float Elds[2][EROWS * LDS_STRIDE];     // double-buffered embedding stage
    __shared__ float xt[WAVES * 16 * LDS_STRIDE];     // per-wave 16x64 x tiles
    __shared__ float enlds[K_CODES];                  // |e_k|^2, block-resident
    __shared__ int   wcodes[WAVES * 16];

    const int tid  = threadIdx.x;
    const int wave = tid >> 5;
    const int lane = tid & 31;
    const int lrow = lane & 15;   // N column within chunk / M sub-row
    const int half = lane >> 4;   // lane group 0/1

    const int  tile = blockIdx.x * WAVES + wave;
    const long m0   = (long)tile * 16;

    float* myxt = &xt[wave * 16 * LDS_STRIDE];

    // ---- async-stage E chunk s into Elds[s&1] (ASYNCcnt path, bypasses VGPRs) ----
    auto stage_async = [&](int s) {
        const int p = s & 1;
#pragma unroll
        for (int i = 0; i < 4; ++i) {                 // 64*16/256 = 4 float4 per thread
            int f   = tid + 256 * i;
            int row = f >> 4;
            int c4  = f & 15;
            unsigned lds = (unsigned)(size_t)(void*)&Elds[p][row * LDS_STRIDE + c4 * 4];
            unsigned long long ga =
                (unsigned long long)(size_t)&E[(size_t)(s * EROWS + row) * D_DIM + c4 * 4];
            asm volatile("global_load_async_to_lds_b128 %0, %1, off"
                         :: "v"(lds), "v"(ga) : "memory");
        }
    };

    stage_async(0);   // overlap first E chunk with x/enorm staging

    // ---- stage this wave's 16x64 x tile into LDS ----
#pragma unroll
    for (int i = 0; i < 8; ++i) {
        int f   = lane + 32 * i;
        int row = f >> 4;
        int c4  = f & 15;
        float4 v = *(const float4*)&x[(m0 + row) * D_DIM + c4 * 4];
        *(float4*)&myxt[row * LDS_STRIDE + c4 * 4] = v;
    }
    // ---- stage enorm (4KB) into LDS ----
    *(float4*)&enlds[tid * 4] = *(const float4*)&enorm[tid * 4];
    __syncthreads();

    // ---- preload A fragments: lane holds x[lrow][4c + 2*half .. +1] ----
    v2f afrag[16];
#pragma unroll
    for (int c = 0; c < 16; ++c)
        afrag[c] = *(const v2f*)&myxt[lrow * LDS_STRIDE + c * 4 + 2 * half];

    float minv[8];
    int   mini[8];
#pragma unroll
    for (int j = 0; j < 8; ++j) { minv[j] = 3.4e38f; mini[j] = 0; }

    // ---- pipelined sweep over all K codes ----
    for (int s = 0; s < NSTAGES; ++s) {
        asm volatile("s_wait_asynccnt 0x0" ::: "memory");  // buffer s&1 landed (this wave's issues)
        __syncthreads();                                   // all waves: staged + prev buffer free
        if (s + 1 < NSTAGES) stage_async(s + 1);           // prefetch next chunk into other buffer

        const float* Eb = Elds[s & 1];
#pragma unroll
        for (int t = 0; t < EROWS / 16; ++t) {
            const int   n0 = s * EROWS + t * 16;
            const float en = enlds[n0 + lrow];             // ds_load, no VMEM stall
            v8f acc = {};
            const float* eb = &Eb[(t * 16 + lrow) * LDS_STRIDE + 2 * half];
#pragma unroll
            for (int c = 0; c < 16; ++c) {                 // K-dim: D=64 in steps of 4
                v2f bfrag = *(const v2f*)&eb[c * 4];
                acc = __builtin_amdgcn_wmma_f32_16x16x4_f32(
                        false, afrag[c], false, bfrag, (short)0, acc, false, false);
            }
            const int kk = n0 + lrow;
#pragma unroll
            for (int j = 0; j < 8; ++j) {                  // row M = j + 8*half
                float dist = en - 2.0f * acc[j];
                if (dist < minv[j]) { minv[j] = dist; mini[j] = kk; }
            }
        }
    }

    // ---- argmin across the 16 lanes of each half, lowest-index tiebreak ----
#pragma unroll
    for (int j = 0; j < 8; ++j) {
        float v = minv[j]; int idx = mini[j];
#pragma unroll
        for (int off = 1; off < 16; off <<= 1) {
            float ov = __shfl_xor(v, off, 32);
            int   oi = __shfl_xor(idx, off, 32);
            if (ov < v || (ov == v && oi < idx)) { v = ov; idx = oi; }
        }
        minv[j] = v; mini[j] = idx;
    }
    if (lrow == 0) {
#pragma unroll
        for (int j = 0; j < 8; ++j) wcodes[wave * 16 + j + 8 * half] = mini[j];
    }
    __syncthreads();

    // ---- epilogue: codes, cluster-size scatter, quantized/residuals, loss, embed-sum scatter ----
    if (lane < 16) {
        int c = wcodes[wave * 16 + lane];
        out_codes[m0 + lane] = (float)c;
        atomicAdd(&ws_cluster[c], 1.0f);
    }
    float lsum = 0.0f;
    for (int r = 0; r < 16; ++r) {
        const int  c      = wcodes[wave * 16 + r];
        const long rowoff = (m0 + r) * D_DIM;
#pragma unroll
        for (int h = 0; h < 2; ++h) {
            const int   d  = lane + 32 * h;
            const float e  = E[(size_t)c * D_DIM + d];
            const float xv = myxt[r * LDS_STRIDE + d];
            out_q[rowoff + d]   = e;              // x + sg(e - x) == e
            const float res = xv - e;
            out_res[rowoff + d] = res;
            lsum += res * res;
            atomicAdd(&ws_esum[(size_t)c * D_DIM + d], xv);
        }
    }
#pragma unroll
    for (int off = 16; off >= 1; off >>= 1) lsum += __shfl_xor(lsum, off, 32);
    if (lane == 0) atomicAdd(ws_loss, lsum);
}

// ---------------- EMA finalize ----------------
__global__ void vq_finalize(const float* __restrict__ ema_cs, const float* __restrict__ ema_es,
                            const float* __restrict__ ws_cluster, const float* __restrict__ ws_esum,
                            const float* __restrict__ ws_loss,
                            float* __restrict__ out_loss, float* __restrict__ out_nemb,
                            float* __restrict__ out_ncs, float* __restrict__ out_nes)
{
    int kd = blockIdx.x * blockDim.x + threadIdx.x;   // 0 .. K*D-1
    if (kd >= K_CODES * D_DIM) return;
    int k = kd >> 6;
    float ncs = DECAY_W * ema_cs[k]  + (1.0f - DECAY_W) * ws_cluster[k];
    float nes = DECAY_W * ema_es[kd] + (1.0f - DECAY_W) * ws_esum[kd];
    out_nes[kd]  = nes;
    out_nemb[kd] = nes / fmaxf(ncs, EPS_W);
    if ((kd & 63) == 0) out_ncs[k] = ncs;
    if (kd == 0)
        out_loss[0] = (1.0f + BETA_W) * ws_loss[0] / (float)(B_TOTAL * D_DIM);
}

extern "C" void kernel_launch(void* const* d_in, const int* in_sizes, int n_in,
                              void* d_out, int out_size, void* d_ws, size_t ws_size,
                              hipStream_t stream) {
    const float* x       = (const float*)d_in[0];   // [B, D]
    const float* E       = (const float*)d_in[1];   // [K, D]
    const float* ema_cs  = (const float*)d_in[2];   // [K]
    const float* ema_es  = (const float*)d_in[3];   // [K, D]

    // workspace layout (floats)
    float* ws        = (float*)d_ws;
    float* ws_enorm  = ws;                                 // K
    float* ws_clust  = ws + K_CODES;                       // K
    float* ws_esum   = ws + 2 * K_CODES;                   // K*D
    float* ws_loss   = ws + 2 * K_CODES + K_CODES * D_DIM; // 1

    // output layout (floats), reference return order
    float* out      = (float*)d_out;
    float* o_codes  = out;                               // B
    float* o_q      = o_codes + B_TOTAL;                 // B*D
    float* o_res    = o_q + (size_t)B_TOTAL * D_DIM;     // B*D
    float* o_loss   = o_res + (size_t)B_TOTAL * D_DIM;   // 1
    float* o_nemb   = o_loss + 1;                        // K*D
    float* o_ncs    = o_nemb + K_CODES * D_DIM;          // K
    float* o_nes    = o_ncs + K_CODES;                   // K*D

    // 1) zero accumulators (cluster + esum + loss)
    int zcount = K_CODES + K_CODES * D_DIM + 1;
    vq_zero<<<(zcount + 255) / 256, 256, 0, stream>>>(ws_clust, zcount);

    // 2) |e_k|^2
    vq_enorm<<<(K_CODES + 255) / 256, 256, 0, stream>>>(E, ws_enorm);

    // 3) fused distance-GEMM / argmin / quantize / scatter
    int tiles  = B_TOTAL / 16;          // 16384
    int blocks = tiles / WAVES;         // 2048
    vq_main<<<blocks, 256, 0, stream>>>(x, E, ws_enorm, o_codes, o_q, o_res,
                                        ws_clust, ws_esum, ws_loss);

    // 4) EMA update + loss
    vq_finalize<<<(K_CODES * D_DIM + 255) / 256, 256, 0, stream>>>(
        ema_cs, ema_es, ws_clust, ws_esum, ws_loss, o_loss, o_nemb, o_ncs, o_nes);
}